// AR_14328010899741
// MI455X (gfx1250) — compile-verified
//
#include <hip/hip_runtime.h>

typedef __attribute__((ext_vector_type(2))) float v2f;
typedef __attribute__((ext_vector_type(8))) float v8f;

#define BATCH   256
#define T_SEQ   168
#define CHN     1024
#define HORIZON 24
#define MPAD    32   // A padded to 32 rows -> two 16-row WMMA tiles

// ---------------------------------------------------------------------------
// Kernel 1: collapse the sequential 24-step AR rollout into an affine map
//   pred_h = beta_h + sum_j A[h][j] * y[j]
// A is 24x168 (padded to 32x168 with zero rows), beta is 24 (padded to 32).
// Single workgroup; trivial cost (~24*168*24 MACs).
// ---------------------------------------------------------------------------
__global__ __launch_bounds__(256) void build_coeffs(const float* __restrict__ w,
                                                    const float* __restrict__ b,
                                                    float* __restrict__ A,
                                                    float* __restrict__ beta) {
    __shared__ float sw[T_SEQ];
    __shared__ float sA[HORIZON][T_SEQ];
    __shared__ float sbeta[HORIZON];
    const int t = threadIdx.x;
    if (t < T_SEQ) sw[t] = w[t];
    __syncthreads();
    for (int h = 0; h < HORIZON; ++h) {
        if (t < T_SEQ) {
            float v = (t >= h) ? sw[t - h] : 0.0f;
            for (int k = 0; k < h; ++k) v += sw[k + T_SEQ - h] * sA[k][t];
            sA[h][t] = v;
        }
        if (t == 0) {
            float bv = b[0];
            for (int k = 0; k < h; ++k) bv += sw[k + T_SEQ - h] * sbeta[k];
            sbeta[h] = bv;
        }
        __syncthreads();
    }
    for (int idx = t; idx < MPAD * T_SEQ; idx += 256) {
        int r = idx / T_SEQ, c = idx - r * T_SEQ;
        A[idx] = (r < HORIZON) ? sA[r][c] : 0.0f;
    }
    if (t < MPAD) beta[t] = (t < HORIZON) ? sbeta[t] : 0.0f;
}

// ---------------------------------------------------------------------------
// Kernel 2: batched GEMM  out[b] = A(32x168) @ Y[b](168x1024) + beta
// via V_WMMA_F32_16X16X4_F32. One wave owns one (batch, 16-col) tile and
// accumulates both 16-row M-tiles against the same B fragment, so Y (176 MB)
// is streamed from HBM exactly once -> memory-bound at ~8.6 us.
//
// Fragment layouts (ISA 7.12.2, 32-bit):
//   A 16x4 : lane m = lane&15; VGPR0/1 hold K = k0+2*half, k0+2*half+1
//   B 4x16 : lane n = lane&15; VGPR0/1 hold K-rows k0+2*half, k0+2*half+1
//   C 16x16: VGPR r -> row r + 8*half, col = lane&15
// ---------------------------------------------------------------------------
__global__ __launch_bounds__(256) void ar_gemm(const float* __restrict__ Y,
                                               const float* __restrict__ A,
                                               const float* __restrict__ beta,
                                               float* __restrict__ out) {
    const int lane = threadIdx.x & 31;
    const int wave = blockIdx.x * 8 + (threadIdx.x >> 5);
    const int nTile = wave & 63;     // 64 tiles of 16 channels
    const int batch = wave >> 6;     // 256 batches
    const int col   = (nTile << 4) + (lane & 15);
    const int half  = lane >> 4;
    const int m     = lane & 15;

    const float* __restrict__ Yb = Y + (size_t)batch * (size_t)(T_SEQ * CHN);

    v8f acc0 = {};
    v8f acc1 = {};

    for (int k0 = 0; k0 < T_SEQ; k0 += 4) {
        const int kk = k0 + 2 * half;
        v2f a0 = *(const v2f*)(A + m * T_SEQ + kk);            // rows 0..15
        v2f a1 = *(const v2f*)(A + (16 + m) * T_SEQ + kk);     // rows 16..31
        v2f bf;
        bf.x = Yb[(size_t)kk * CHN + col];
        bf.y = Yb[(size_t)(kk + 1) * CHN + col];
        acc0 = __builtin_amdgcn_wmma_f32_16x16x4_f32(false, a0, false, bf,
                                                     (short)0, acc0, false, false);
        acc1 = __builtin_amdgcn_wmma_f32_16x16x4_f32(false, a1, false, bf,
                                                     (short)0, acc1, false, false);
    }

    float* __restrict__ ob = out + (size_t)batch * (size_t)(HORIZON * CHN);
#pragma unroll
    for (int r = 0; r < 8; ++r) {
        const int m0 = r + 8 * half;                 // rows 0..15: all valid
        ob[(size_t)m0 * CHN + col] = acc0[r] + beta[m0];
    }
    if (half == 0) {                                 // rows 16..23 valid, 24..31 are pad
#pragma unroll
        for (int r = 0; r < 8; ++r) {
            const int m1 = 16 + r;
            ob[(size_t)m1 * CHN + col] = acc1[r] + beta[m1];
        }
    }
}

// ---------------------------------------------------------------------------
extern "C" void kernel_launch(void* const* d_in, const int* in_sizes, int n_in,
                              void* d_out, int out_size, void* d_ws, size_t ws_size,
                              hipStream_t stream) {
    // setup_inputs order: x (unused), y, w, b
    const float* y = (const float*)d_in[1];
    const float* w = (const float*)d_in[2];
    const float* b = (const float*)d_in[3];

    float* A    = (float*)d_ws;          // 32*168 floats
    float* beta = A + MPAD * T_SEQ;      // 32 floats

    build_coeffs<<<1, 256, 0, stream>>>(w, b, A, beta);

    const int totalWaves = BATCH * (CHN / 16);       // 16384
    ar_gemm<<<totalWaves / 8, 256, 0, stream>>>(y, A, beta, (float*)d_out);
}